// DynamicKernelBlock_46162308498099
// MI455X (gfx1250) — compile-verified
//
#include <hip/hip_runtime.h>
#include <math.h>

#define B_ 32
#define C_ 64
#define L_ 4096
#define NFFT 4096
#define PHYS_N 12
#define LT 64     // l-tile per workgroup in K8
#define XW 128    // staged x window width (LT + 32 halo each side)
#define XSTR 132  // padded LDS stride for x stage (132 % 64 = 4 -> conflict-free-ish)
#define ASTR 68   // padded LDS stride for transposed act tile

typedef __attribute__((ext_vector_type(2))) float v2f;
typedef __attribute__((ext_vector_type(8))) float v8f;

// ---- workspace layout (float offsets) ----
#define WS_XM    0                         // B*C
#define WS_JMAX  (WS_XM + B_*C_)           // B*C
#define WS_JRMAX (WS_JMAX + B_*C_)         // B*C
#define WS_ZC    (WS_JRMAX + B_*C_)        // B*C
#define WS_SVM   (WS_ZC + B_*C_)           // B*L
#define WS_PHYS  (WS_SVM + B_*L_)          // B*12
#define WS_WGT   (WS_PHYS + B_*PHYS_N)     // B*4
#define WS_MU    (WS_WGT + B_*4)           // B
#define WS_INV   (WS_MU + B_)              // B

// ---------------- block reduction helpers (blockDim.x == 256) ----------------
__device__ __forceinline__ float blk_reduce_sum(float v, float* s) {
  int t = threadIdx.x;
  s[t] = v; __syncthreads();
  for (int o = 128; o > 0; o >>= 1) { if (t < o) s[t] += s[t + o]; __syncthreads(); }
  float r = s[0]; __syncthreads();
  return r;
}
__device__ __forceinline__ float blk_reduce_max(float v, float* s) {
  int t = threadIdx.x;
  s[t] = v; __syncthreads();
  for (int o = 128; o > 0; o >>= 1) { if (t < o) s[t] = fmaxf(s[t], s[t + o]); __syncthreads(); }
  float r = s[0]; __syncthreads();
  return r;
}

// ---------------- K1: per-(b,c) mean, jerk stats, zero crossings ----------------
__global__ void k1_chan_stats(const float* __restrict__ x, float* __restrict__ ws) {
  __shared__ float s[256];
  int bc = blockIdx.x;               // b*C + c
  const float* xr = x + (size_t)bc * L_;
  int t = threadIdx.x;
  float sum = 0.f, jm = 0.f, jrm = 0.f, zc = 0.f;
  for (int l = t; l < L_; l += 256) {
    float x0 = xr[l];
    sum += x0;
    if (l < L_ - 1) {
      float x1 = xr[l + 1];
      float j = (x1 - x0) * 50.0f;
      jm = fmaxf(jm, fabsf(j));
      if (x0 * x1 < 0.f) zc += 1.f;
      if (l < L_ - 2) {
        float x2 = xr[l + 2];
        float jr = (x2 - 2.f * x1 + x0) * 2500.0f;   // SR*SR
        jrm = fmaxf(jrm, fabsf(jr));
      }
    }
  }
  float S   = blk_reduce_sum(sum, s);
  float JM  = blk_reduce_max(jm, s);
  float JRM = blk_reduce_max(jrm, s);
  float ZC  = blk_reduce_sum(zc, s);
  if (t == 0) {
    ws[WS_XM + bc]    = S / (float)L_;
    ws[WS_JMAX + bc]  = JM;
    ws[WS_JRMAX + bc] = JRM;
    ws[WS_ZC + bc]    = ZC;
  }
}

// ---------------- K2: svm[b,l] = sqrt(sum_c x^2) ----------------
__global__ void k2_svm(const float* __restrict__ x, float* __restrict__ ws) {
  int idx = blockIdx.x * 256 + threadIdx.x;   // b*L + l
  if (idx >= B_ * L_) return;
  int b = idx >> 12, l = idx & (L_ - 1);
  const float* xb = x + (size_t)b * C_ * L_ + l;
  float s = 0.f;
  for (int c = 0; c < C_; c++) { float v = xb[(size_t)c * L_]; s += v * v; }
  ws[WS_SVM + idx] = sqrtf(s);
}

// ---------------- K3: per-b svm statistics ----------------
__global__ void k3_svm_stats(float* __restrict__ ws) {
  __shared__ float s[256];
  __shared__ float sh_mean, sh_std, sh_max;
  int b = blockIdx.x, t = threadIdx.x;
  const float* svm = ws + WS_SVM + (size_t)b * L_;
  float sum = 0.f, sq = 0.f, mx = -1e30f;
  for (int l = t; l < L_; l += 256) { float v = svm[l]; sum += v; sq += v * v; mx = fmaxf(mx, v); }
  float S = blk_reduce_sum(sum, s);
  float Q = blk_reduce_sum(sq, s);
  float M = blk_reduce_max(mx, s);
  if (t == 0) {
    sh_mean = S / (float)L_;
    float var = (Q - S * S / (float)L_) / (float)(L_ - 1);   // ddof=1
    sh_std = sqrtf(fmaxf(var, 0.f)) + 1e-6f;
    sh_max = M;
  }
  __syncthreads();
  float mean = sh_mean, std = sh_std;
  float thr = mean + 2.f * std, hthr = mean + 3.f * std, low = mean - std;
  const int third = 1365;
  float c1 = 0.f, c2 = 0.f, c3 = 0.f, s2 = 0.f, q2 = 0.f, pre = 0.f, post = 0.f, mid = -1e30f;
  for (int l = t; l < L_; l += 256) {
    float v = svm[l];
    if (v > thr)  c1 += 1.f;
    if (v > hthr) c2 += 1.f;
    if (v < low)  c3 += 1.f;
    if (l >= L_ / 2) { s2 += v; q2 += v * v; }
    if (l < third)            pre  += v;
    else if (l < 2 * third)   mid = fmaxf(mid, v);
    else                      post += v;
  }
  c1 = blk_reduce_sum(c1, s); c2 = blk_reduce_sum(c2, s); c3 = blk_reduce_sum(c3, s);
  s2 = blk_reduce_sum(s2, s); q2 = blk_reduce_sum(q2, s);
  pre = blk_reduce_sum(pre, s); post = blk_reduce_sum(post, s);
  mid = blk_reduce_max(mid, s);
  if (t == 0) {
    float* ph = ws + WS_PHYS + b * PHYS_N;
    ph[0] = sh_max;
    ph[1] = mean;
    ph[5] = c1 / (float)L_;    // impact_duration
    ph[6] = c2 / (float)L_;    // impact_ratio
    ph[8] = c3 / (float)L_;    // free_fall
    const int n2 = L_ / 2;
    float var2 = (q2 - s2 * s2 / (float)n2) / (float)(n2 - 1);
    float std2 = sqrtf(fmaxf(var2, 0.f)) + 1e-6f;
    ph[7] = tanhf(1.0f / std2);                       // post_stillness
    float prem  = pre  / (float)third;
    float postm = post / (float)(L_ - 2 * third);
    float fp = fmaxf(mid - prem, 0.f) + fmaxf(mid - postm, 0.f);
    ph[11] = fp / (fabsf(mid) + 1e-6f);               // fall_phase
  }
}

// ---------------- K4: per-b rFFT power spectrum (radix-2 DIT in LDS) ----------------
__global__ void k4_fft(const float* __restrict__ x, float* __restrict__ ws) {
  __shared__ float re[NFFT];
  __shared__ float im[NFFT];
  __shared__ float pw[NFFT / 2 + 1];
  __shared__ float s[256];
  int b = blockIdx.x, t = threadIdx.x;
  for (int f = t; f <= NFFT / 2; f += 256) pw[f] = 0.f;
  __syncthreads();
  for (int c = 0; c < C_; c++) {
    const float* xr = x + ((size_t)b * C_ + c) * L_;
    for (int i = t; i < NFFT; i += 256) {
      unsigned r = __brev((unsigned)i) >> 20;     // 12-bit reversal
      re[r] = xr[i];
      im[r] = 0.f;
    }
    __syncthreads();
    for (int len = 2; len <= NFFT; len <<= 1) {
      int half = len >> 1;
      float base = -6.283185307179586f / (float)len;
      for (int k = t; k < NFFT / 2; k += 256) {
        int grp = k / half, j = k - grp * half;
        int i0 = grp * len + j, i1 = i0 + half;
        float wr, wi;
        __sincosf(base * (float)j, &wi, &wr);     // wi = sin, wr = cos
        float ur = re[i0], ui = im[i0];
        float xr1 = re[i1], xi1 = im[i1];
        float vr = xr1 * wr - xi1 * wi;
        float vi = xr1 * wi + xi1 * wr;
        re[i0] = ur + vr; im[i0] = ui + vi;
        re[i1] = ur - vr; im[i1] = ui - vi;
      }
      __syncthreads();
    }
    const float inv = 1.0f / (float)NFFT;         // ortho norm -> |X|^2 / N
    for (int f = t; f <= NFFT / 2; f += 256)
      pw[f] += (re[f] * re[f] + im[f] * im[f]) * inv;
    __syncthreads();
  }
  float tot = 0.f, cen = 0.f, hfr = 0.f;
  for (int f = t; f <= NFFT / 2; f += 256) {
    float p = pw[f];
    tot += p;
    cen += p * ((float)f / (float)(NFFT / 2));
    if (f >= 820) hfr += p;                       // freqs[f] = f/2048 > 0.4
  }
  tot = blk_reduce_sum(tot, s);
  cen = blk_reduce_sum(cen, s);
  hfr = blk_reduce_sum(hfr, s);
  if (t == 0) {
    float* ph = ws + WS_PHYS + b * PHYS_N;
    float tp = tot + 1e-6f;
    ph[9]  = cen / tp;   // spectral_centroid
    ph[10] = hfr / tp;   // high_freq_ratio
  }
}

// ---------------- K6: gating MLPs -> branch weights ----------------
__global__ void k6_gating(const float* __restrict__ comp_w1, const float* __restrict__ comp_b1,
                          const float* __restrict__ comp_w2, const float* __restrict__ comp_b2,
                          const float* __restrict__ pn_g, const float* __restrict__ pn_b,
                          const float* __restrict__ pn_m, const float* __restrict__ pn_v,
                          const float* __restrict__ pe_w1, const float* __restrict__ pe_b1,
                          const float* __restrict__ pe_w2, const float* __restrict__ pe_b2,
                          const float* __restrict__ pg_w, const float* __restrict__ pg_b,
                          const float* __restrict__ temperature,
                          float* __restrict__ ws) {
  int b = threadIdx.x;
  if (b >= B_) return;
  float p[PHYS_N];
  const float* ph = ws + WS_PHYS + b * PHYS_N;
  float jm = 0.f, jrm = 0.f, zc = 0.f;
  for (int c = 0; c < C_; c++) {
    jm  += ws[WS_JMAX  + b * C_ + c];
    jrm += ws[WS_JRMAX + b * C_ + c];
    zc  += ws[WS_ZC    + b * C_ + c];
  }
  p[0] = ph[0]; p[1] = ph[1];
  p[2] = jm / (float)C_;
  p[3] = jrm / (float)C_;
  p[4] = (zc / (float)C_) / (float)L_;
  p[5] = ph[5]; p[6] = ph[6]; p[7] = ph[7]; p[8] = ph[8];
  p[9] = ph[9]; p[10] = ph[10]; p[11] = ph[11];
  for (int i = 0; i < PHYS_N; i++)
    if (!(p[i] == p[i]) || fabsf(p[i]) > 3.0e38f) p[i] = 0.f;   // nan_to_num
  float pn[PHYS_N];
  for (int i = 0; i < PHYS_N; i++)
    pn[i] = (p[i] - pn_m[i]) * rsqrtf(pn_v[i] + 1e-5f) * pn_g[i] + pn_b[i];
  float h1[32];
  for (int j = 0; j < 32; j++) {
    float a = pe_b1[j];
    for (int i = 0; i < PHYS_N; i++) a += pn[i] * pe_w1[i * 32 + j];
    h1[j] = fmaxf(a, 0.f);
  }
  float h2[32];
  for (int k = 0; k < 32; k++) {
    float a = pe_b2[k];
    for (int j = 0; j < 32; j++) a += h1[j] * pe_w2[j * 32 + k];
    h2[k] = fmaxf(a, 0.f);
  }
  float pwg[4];
  for (int q = 0; q < 4; q++) {
    float a = pg_b[q];
    for (int k = 0; k < 32; k++) a += h2[k] * pg_w[k * 4 + q];
    pwg[q] = 1.0f / (1.0f + expf(-a));
  }
  float h[64];
  const float* xm = ws + WS_XM + b * C_;
  for (int j = 0; j < 64; j++) {
    float a = comp_b1[j];
    for (int c = 0; c < C_; c++) a += xm[c] * comp_w1[c * 64 + j];
    h[j] = fmaxf(a, 0.f);
  }
  float z[4], zmax = -1e30f;
  float temp = fmaxf(temperature[0], 0.1f);
  for (int q = 0; q < 4; q++) {
    float a = comp_b2[q];
    for (int j = 0; j < 64; j++) a += h[j] * comp_w2[j * 4 + q];
    z[q] = a * pwg[q] / temp;
    zmax = fmaxf(zmax, z[q]);
  }
  float den = 0.f;
  for (int q = 0; q < 4; q++) { z[q] = expf(z[q] - zmax); den += z[q]; }
  for (int q = 0; q < 4; q++) ws[WS_WGT + b * 4 + q] = z[q] / den;
}

// ---------------- fused conv4 + BN + mix at one (b,c,l) (global-memory variant) ----------------
__device__ __forceinline__ float mixed_val(const float* __restrict__ x,
    const float* __restrict__ w7,  const float* __restrict__ w15,
    const float* __restrict__ w31, const float* __restrict__ w63,
    const float* __restrict__ bn_g, const float* __restrict__ bn_b,
    const float* __restrict__ bn_m, const float* __restrict__ bn_v,
    const float* __restrict__ wgt, int b, int c, int l) {
  const float* xr = x + ((size_t)b * C_ + c) * L_;
  const int ks[4] = {7, 15, 31, 63};
  const float* wp[4] = {w7 + c * 7, w15 + c * 15, w31 + c * 31, w63 + c * 63};
  float out = 0.f;
  #pragma unroll
  for (int i = 0; i < 4; i++) {
    int k = ks[i], pp = (k - 1) >> 1;
    float acc = 0.f;
    const float* wr = wp[i];
    for (int t = 0; t < k; t++) {
      int ll = l - pp + t;
      if (ll >= 0 && ll < L_) acc += xr[ll] * wr[t];
    }
    float sc = bn_g[i * C_ + c] * rsqrtf(bn_v[i * C_ + c] + 1e-5f);
    float y = (acc - bn_m[i * C_ + c]) * sc + bn_b[i * C_ + c];
    out += wgt[i] * y;
  }
  return out;
}

// ---------------- K7: per-b GroupNorm statistics over recomputed mix ----------------
__global__ void k7_gn_stats(const float* __restrict__ x,
    const float* __restrict__ w7,  const float* __restrict__ w15,
    const float* __restrict__ w31, const float* __restrict__ w63,
    const float* __restrict__ bn_g, const float* __restrict__ bn_b,
    const float* __restrict__ bn_m, const float* __restrict__ bn_v,
    float* __restrict__ ws) {
  __shared__ float s[256];
  __shared__ float wl[4];
  int b = blockIdx.x, t = threadIdx.x;
  if (t < 4) wl[t] = ws[WS_WGT + b * 4 + t];
  __syncthreads();
  float sum = 0.f, sq = 0.f;
  for (int idx = t; idx < C_ * L_; idx += 256) {
    int c = idx >> 12, l = idx & (L_ - 1);
    float v = mixed_val(x, w7, w15, w31, w63, bn_g, bn_b, bn_m, bn_v, wl, b, c, l);
    sum += v; sq += v * v;
  }
  float S = blk_reduce_sum(sum, s);
  float Q = blk_reduce_sum(sq, s);
  if (t == 0) {
    const float n = (float)(C_ * L_);
    float mu = S / n;
    float var = Q / n - mu * mu;                 // ddof=0
    ws[WS_MU + b]  = mu;
    ws[WS_INV + b] = rsqrtf(fmaxf(var, 0.f) + 1e-5f);
  }
}

// ---------------- K8: async-stage x -> fused mix/GN/GELU -> WMMA projection ----------------
__global__ void k8_out(const float* __restrict__ x,
    const float* __restrict__ w7,  const float* __restrict__ w15,
    const float* __restrict__ w31, const float* __restrict__ w63,
    const float* __restrict__ bn_g, const float* __restrict__ bn_b,
    const float* __restrict__ bn_m, const float* __restrict__ bn_v,
    const float* __restrict__ gn_g, const float* __restrict__ gn_b,
    const float* __restrict__ proj_w, const float* __restrict__ proj_b,
    const float* __restrict__ res_scale,
    const float* __restrict__ ws, float* __restrict__ out) {
  __shared__ float sX[C_ * XSTR];      // staged x window: [c][j], j = l - (l0-32)
  __shared__ float sActT[LT * ASTR];   // transposed act tile: [l][c]
  __shared__ float wl[4];
  __shared__ float sMu, sInv, sRes;
  int blk = blockIdx.x;
  int b  = blk / (L_ / LT);
  int l0 = (blk % (L_ / LT)) * LT;
  int t = threadIdx.x;
  if (t < 4)  wl[t] = ws[WS_WGT + b * 4 + t];
  if (t == 4) sMu  = ws[WS_MU + b];
  if (t == 5) sInv = ws[WS_INV + b];
  if (t == 6) sRes = res_scale[0];
  // warm proj_w toward the caches (emits global_prefetch_b8)
  __builtin_prefetch(proj_w + t * 16, 0, 0);

  // ---- stage x tile (with 32-element halo each side) into LDS via async loads ----
  unsigned xbase = (unsigned)(uintptr_t)(void*)sX;   // low 32 bits of flat addr = LDS offset
  for (int i = t; i < C_ * XW; i += 256) {
    int c = i >> 7, j = i & (XW - 1);
    int gl = l0 - 32 + j;
    if (gl >= 0 && gl < L_) {
      unsigned loff = xbase + (unsigned)(c * XSTR + j) * 4u;
      unsigned long long gaddr =
          (unsigned long long)(uintptr_t)(x + ((size_t)b * C_ + c) * L_ + gl);
      asm volatile("global_load_async_to_lds_b32 %0, %1, off"
                   :: "v"(loff), "v"(gaddr) : "memory");
    } else {
      sX[c * XSTR + j] = 0.f;          // zero-pad halo (disjoint LDS slots, no hazard)
    }
  }
  asm volatile("s_wait_asynccnt 0" ::: "memory");
  __syncthreads();

  // ---- fused conv4 + BN + mix + GN + exact GELU, act stored transposed ----
  float mu = sMu, inv = sInv;
  {
    int c  = t & 63;
    int lo0 = t >> 6;                  // 16 l-rows per thread (lo0, lo0+4, ...)
    const int ks[4] = {7, 15, 31, 63};
    const float* wp[4] = {w7 + c * 7, w15 + c * 15, w31 + c * 31, w63 + c * 63};
    float sc[4], of[4];
    #pragma unroll
    for (int i = 0; i < 4; i++) {
      float scale = bn_g[i * C_ + c] * rsqrtf(bn_v[i * C_ + c] + 1e-5f);
      sc[i] = wl[i] * scale;
      of[i] = wl[i] * (bn_b[i * C_ + c] - bn_m[i * C_ + c] * scale);
    }
    for (int lo = lo0; lo < LT; lo += 4) {
      const float* xr = &sX[c * XSTR + 32 + lo];   // centered at tap offset 0
      float v = of[0] + of[1] + of[2] + of[3];
      #pragma unroll
      for (int i = 0; i < 4; i++) {
        int k = ks[i], pp = (k - 1) >> 1;
        float acc = 0.f;
        for (int tt = 0; tt < k; tt++) acc += xr[tt - pp] * wp[i][tt];
        v += sc[i] * acc;
      }
      float g = (v - mu) * inv * gn_g[c] + gn_b[c];
      float a = 0.5f * g * (1.0f + erff(g * 0.70710678118654752f));   // exact GELU
      sActT[lo * ASTR + c] = a;
    }
  }
  __syncthreads();

  // ---- GEMM: out[o,l] = sum_c proj_w[o,c]*act[c,l]; 8 waves x 2 tiles of 16x16 ----
  int wave = t >> 5;
  int lane = t & 31;
  int half = lane >> 4;                // 0: lanes 0-15, 1: lanes 16-31
  int lan  = lane & 15;
  float rs = sRes;
  for (int ti = wave; ti < 16; ti += 8) {
    int m0 = (ti >> 2) * 16;
    int n0 = (ti & 3) * 16;
    v8f acc = {0.f, 0.f, 0.f, 0.f, 0.f, 0.f, 0.f, 0.f};
    for (int k0 = 0; k0 < C_; k0 += 4) {
      int ka = k0 + 2 * half;
      // A 16x4: lanes 0-15 hold {K0,K1}, lanes 16-31 hold {K2,K3}; contiguous -> b64 load
      v2f A  = *(const v2f*)&proj_w[(m0 + lan) * C_ + ka];
      // B 4x16: rows striped across lanes; transposed act makes {Kka,Kka+1} contiguous
      v2f Bv = *(const v2f*)&sActT[(n0 + lan) * ASTR + ka];
      acc = __builtin_amdgcn_wmma_f32_16x16x4_f32(
          /*neg_a=*/false, A, /*neg_b=*/false, Bv,
          /*c_mod=*/(short)0, acc, /*reuse_a=*/false, /*reuse_b=*/false);
    }
    int n = n0 + lan;
    #pragma unroll
    for (int r = 0; r < 8; r++) {
      int m = m0 + r + 8 * half;       // D layout: VGPR r holds rows r / r+8
      size_t oi = ((size_t)b * C_ + m) * L_ + l0 + n;
      float v = acc[r] + proj_b[m];
      out[oi] = x[oi] + rs * v;
    }
  }
}

// ---------------- host launcher ----------------
extern "C" void kernel_launch(void* const* d_in, const int* in_sizes, int n_in,
                              void* d_out, int out_size, void* d_ws, size_t ws_size,
                              hipStream_t stream) {
  const float* x       = (const float*)d_in[0];
  const float* w7      = (const float*)d_in[1];
  const float* w15     = (const float*)d_in[2];
  const float* w31     = (const float*)d_in[3];
  const float* w63     = (const float*)d_in[4];
  const float* bn_g    = (const float*)d_in[5];
  const float* bn_b    = (const float*)d_in[6];
  const float* bn_m    = (const float*)d_in[7];
  const float* bn_v    = (const float*)d_in[8];
  const float* comp_w1 = (const float*)d_in[9];
  const float* comp_b1 = (const float*)d_in[10];
  const float* comp_w2 = (const float*)d_in[11];
  const float* comp_b2 = (const float*)d_in[12];
  const float* pn_g    = (const float*)d_in[13];
  const float* pn_b    = (const float*)d_in[14];
  const float* pn_m    = (const float*)d_in[15];
  const float* pn_v    = (const float*)d_in[16];
  const float* pe_w1   = (const float*)d_in[17];
  const float* pe_b1   = (const float*)d_in[18];
  const float* pe_w2   = (const float*)d_in[19];
  const float* pe_b2   = (const float*)d_in[20];
  const float* pg_w    = (const float*)d_in[21];
  const float* pg_b    = (const float*)d_in[22];
  const float* temp    = (const float*)d_in[23];
  const float* gn_g    = (const float*)d_in[24];
  const float* gn_b    = (const float*)d_in[25];
  const float* proj_w  = (const float*)d_in[26];
  const float* proj_b  = (const float*)d_in[27];
  const float* res_sc  = (const float*)d_in[28];
  float* out = (float*)d_out;
  float* ws  = (float*)d_ws;
  (void)in_sizes; (void)n_in; (void)out_size; (void)ws_size;

  k1_chan_stats<<<B_ * C_, 256, 0, stream>>>(x, ws);
  k2_svm<<<(B_ * L_ + 255) / 256, 256, 0, stream>>>(x, ws);
  k3_svm_stats<<<B_, 256, 0, stream>>>(ws);
  k4_fft<<<B_, 256, 0, stream>>>(x, ws);
  k6_gating<<<1, 64, 0, stream>>>(comp_w1, comp_b1, comp_w2, comp_b2,
      pn_g, pn_b, pn_m, pn_v, pe_w1, pe_b1, pe_w2, pe_b2, pg_w, pg_b, temp, ws);
  k7_gn_stats<<<B_, 256, 0, stream>>>(x, w7, w15, w31, w63, bn_g, bn_b, bn_m, bn_v, ws);
  k8_out<<<B_ * (L_ / LT), 256, 0, stream>>>(x, w7, w15, w31, w63,
      bn_g, bn_b, bn_m, bn_v, gn_g, gn_b, proj_w, proj_b, res_sc, ws, out);
}